// BboxLoss_40003325395660
// MI455X (gfx1250) — compile-verified
//
#include <hip/hip_runtime.h>
#include <stdint.h>

#define EPS    1e-7f
#define BATCH  64
#define NPRED  2048
#define NTGT   256
#define TTILE  8       // targets per block tile
#define CTHREADS 256
#define GTHREADS 1024

// Builtin signature (from hipcc diagnostic): pointers to int4 vectors.
typedef int v4i_t __attribute__((vector_size(4 * sizeof(int))));
typedef __attribute__((address_space(1))) v4i_t glb_v4i_t;
typedef __attribute__((address_space(3))) v4i_t lds_v4i_t;

// ---- CDNA5 async global -> LDS DMA (ASYNCcnt path) ----------------------
__device__ __forceinline__ void async_copy_b128(const void* g, void* l) {
#if __has_builtin(__builtin_amdgcn_global_load_async_to_lds_b128)
    __builtin_amdgcn_global_load_async_to_lds_b128(
        (glb_v4i_t*)(g), (lds_v4i_t*)(l), /*imm offset*/0, /*cpol*/0);
#else
    *(float4*)l = *(const float4*)g;   // fallback: VGPR-relayed copy
#endif
}

__device__ __forceinline__ void wait_async_lds() {
#if __has_builtin(__builtin_amdgcn_global_load_async_to_lds_b128)
    asm volatile("s_wait_asynccnt 0" ::: "memory");
#endif
}

// ---- Phase 1: cost[t][p] = mean_b (1 - IoU) -----------------------------
__global__ __launch_bounds__(CTHREADS)
void bbox_cost_kernel(const float* __restrict__ pred,
                      const float* __restrict__ tgt,
                      float* __restrict__ cost) {
    // all 64 batches x 8 targets of this tile, staged once: 8 KB
    __shared__ __align__(16) float lds_t[BATCH * TTILE * 4];

    const int tid = threadIdx.x;
    const int p   = blockIdx.x * CTHREADS + tid;
    const int t0  = blockIdx.y * TTILE;

    // DMA the target tile into LDS (512 boxes of 16B; 2 per thread)
    for (int i = tid; i < BATCH * TTILE; i += CTHREADS) {
        const int b = i / TTILE;
        const int j = i % TTILE;
        async_copy_b128(tgt + ((size_t)b * NTGT + t0 + j) * 4, &lds_t[i * 4]);
    }
    wait_async_lds();
    __syncthreads();

    float acc[TTILE];
#pragma unroll
    for (int j = 0; j < TTILE; ++j) acc[j] = 0.0f;

    const float4* pred4 = (const float4*)pred;
    for (int b = 0; b < BATCH; ++b) {
        if (b + 1 < BATCH)
            __builtin_prefetch(&pred4[(size_t)(b + 1) * NPRED + p], 0, 0);
        const float4 pb = pred4[(size_t)b * NPRED + p];
        const float area_p = (pb.z - pb.x) * (pb.w - pb.y);
#pragma unroll
        for (int j = 0; j < TTILE; ++j) {
            const float4 tb = *(const float4*)&lds_t[(b * TTILE + j) * 4];
            const float ix1 = fmaxf(pb.x, tb.x);
            const float iy1 = fmaxf(pb.y, tb.y);
            const float ix2 = fminf(pb.z, tb.z);
            const float iy2 = fminf(pb.w, tb.w);
            const float iw  = fmaxf(ix2 - ix1, 0.0f);
            const float ih  = fmaxf(iy2 - iy1, 0.0f);
            const float inter  = iw * ih;
            const float area_t = (tb.z - tb.x) * (tb.w - tb.y);
            const float uni    = area_p + area_t - inter;
            const float iou    = inter / (uni + EPS);
            acc[j] += 1.0f - iou;
        }
    }

    const float inv_b = 1.0f / (float)BATCH;
#pragma unroll
    for (int j = 0; j < TTILE; ++j)
        cost[(size_t)(t0 + j) * NPRED + p] = acc[j] * inv_b;
}

// ---- Phase 2: sequential greedy matching (single block) -----------------
__global__ __launch_bounds__(GTHREADS)
void greedy_match_kernel(const float* __restrict__ cost,
                         float* __restrict__ out) {
    __shared__ float sval[GTHREADS];
    __shared__ int   sidx[GTHREADS];
    __shared__ unsigned char mask[NPRED];
    __shared__ float ssum;

    const int tid = threadIdx.x;
    for (int i = tid; i < NPRED; i += GTHREADS) mask[i] = 1;
    if (tid == 0) ssum = 0.0f;
    __syncthreads();

    for (int t = 0; t < NTGT; ++t) {
        const float* row = cost + (size_t)t * NPRED;
        float bv = __builtin_inff();
        int   bi = 0;
        for (int i = tid; i < NPRED; i += GTHREADS) {
            const float v = mask[i] ? row[i] : __builtin_inff();
            if (v < bv) { bv = v; bi = i; }   // ascending i => first-occurrence ties
        }
        sval[tid] = bv;
        sidx[tid] = bi;
        __syncthreads();
        for (int s = GTHREADS / 2; s > 0; s >>= 1) {
            if (tid < s) {
                const float ov = sval[tid + s];
                const int   oi = sidx[tid + s];
                if (ov < sval[tid] || (ov == sval[tid] && oi < sidx[tid])) {
                    sval[tid] = ov;
                    sidx[tid] = oi;
                }
            }
            __syncthreads();
        }
        if (tid == 0) {
            ssum += sval[0];
            mask[sidx[0]] = 0;
        }
        __syncthreads();
    }

    if (tid == 0) out[0] = ssum / (float)NTGT;
}

// ---- Launch -------------------------------------------------------------
extern "C" void kernel_launch(void* const* d_in, const int* in_sizes, int n_in,
                              void* d_out, int out_size, void* d_ws, size_t ws_size,
                              hipStream_t stream) {
    (void)in_sizes; (void)n_in; (void)out_size; (void)ws_size;
    const float* pred = (const float*)d_in[0];   // (64, 2048, 4) f32
    const float* tgt  = (const float*)d_in[1];   // (64, 256, 4)  f32
    float* cost = (float*)d_ws;                  // (256, 2048)   f32 = 2 MB

    dim3 grid(NPRED / CTHREADS, NTGT / TTILE);   // (8, 32) blocks
    bbox_cost_kernel<<<grid, CTHREADS, 0, stream>>>(pred, tgt, cost);
    greedy_match_kernel<<<1, GTHREADS, 0, stream>>>(cost, (float*)d_out);
}